// GATConv_G_70428873720562
// MI455X (gfx1250) — compile-verified
//
#include <hip/hip_runtime.h>
#include <hip/hip_bf16.h>
#include <math.h>

#define HH 4
#define FF 32
#define DD 128          // IN = H*F
#define EPSV 1e-6f
#define NSLOPE 0.2f

typedef float v2f __attribute__((ext_vector_type(2)));
typedef float v8f __attribute__((ext_vector_type(8)));

// ---- order-preserving float<->uint encoding for atomic min/max on floats ----
__device__ __forceinline__ unsigned fenc(float f) {
    unsigned u = __float_as_uint(f);
    return (u & 0x80000000u) ? ~u : (u | 0x80000000u);
}
__device__ __forceinline__ float fdec(unsigned u) {
    return __uint_as_float((u & 0x80000000u) ? (u & 0x7FFFFFFFu) : ~u);
}
__device__ __forceinline__ float lrelu(float x) { return x >= 0.f ? x : NSLOPE * x; }
__device__ __forceinline__ float sigmoidf_(float x) { return 1.f / (1.f + __expf(-x)); }
__device__ __forceinline__ float pow_pos(float x, float p) { return __expf(p * __logf(x)); }

// ---------------------------------------------------------------------------
// 0) init accumulators
// ---------------------------------------------------------------------------
__global__ void k_init(float* __restrict__ out, float* __restrict__ denom,
                       unsigned* __restrict__ emax_enc, unsigned* __restrict__ mu_enc, int N) {
    int i = blockIdx.x * blockDim.x + threadIdx.x;
    if (i < N * DD) out[i] = 0.f;
    if (i < N * HH) { denom[i] = 0.f; emax_enc[i] = fenc(-INFINITY); }
    if (i == 0) *mu_enc = fenc(INFINITY);
}

// ---------------------------------------------------------------------------
// 1) feat_proj = feat @ W  via V_WMMA_F32_16X16X4_F32
//    block = 256 threads (8 waves); each block: 4 strips of 16 rows.
//    A tile staged into LDS with CDNA5 async memory->LDS DMA
//    (GLOBAL_LOAD_ASYNC_TO_LDS_B128, ASYNCcnt-tracked, no VGPR round trip).
//    Wave w computes output cols [16w, 16w+16), K=128 in 32 wmma steps.
// ---------------------------------------------------------------------------
__global__ __launch_bounds__(256) void k_gemm_wmma(const float* __restrict__ feat,
                                                   const float* __restrict__ W,
                                                   float* __restrict__ proj, int N) {
    __shared__ float ldsA[16 * DD];        // 8 KB strip of A
    const int lane   = threadIdx.x & 31;   // wave32
    const int laneM  = lane & 15;
    const int laneHi = lane >> 4;          // 0 or 1
    const int kHi    = laneHi * 2;         // K sub-offset 0 / 2
    const int colBase = (threadIdx.x >> 5) * 16;

    // pull the 64 KB weight panel toward L2 (global_prefetch_b8)
    __builtin_prefetch(W + (size_t)threadIdx.x * 64, 0, 1);

    for (int s = 0; s < 4; ++s) {
        const int rowBase = (blockIdx.x * 4 + s) * 16;
        if (rowBase >= N) break;           // uniform across block
        __syncthreads();                   // WAR: previous strip's LDS reads done

        // ---- async DMA the 16x128 f32 A-tile into LDS (2 x b128 per thread)
        const char* gbase = (const char*)(feat + (size_t)rowBase * DD);
        #pragma unroll
        for (int i = threadIdx.x; i < 16 * DD / 4; i += 256) {
            unsigned lds_b = (unsigned)(unsigned long long)(ldsA + i * 4); // LDS byte addr
            unsigned goff  = (unsigned)(i * 16);                           // global byte off
            asm volatile("global_load_async_to_lds_b128 %0, %1, %2 offset:0"
                         :
                         : "v"(lds_b), "v"(goff), "s"(gbase)
                         : "memory");
        }
        asm volatile("s_wait_asynccnt 0x0" ::: "memory");  // own DMAs complete
        __syncthreads();                                   // all waves' tiles visible

        v8f c = {};
        #pragma unroll 4
        for (int k = 0; k < DD; k += 4) {
            v2f a, b;
            // A fragment: lanes 0-15 -> K = {k,k+1}, lanes 16-31 -> K = {k+2,k+3}
            a.x = ldsA[laneM * DD + k + kHi];
            a.y = ldsA[laneM * DD + k + kHi + 1];
            // B fragment (row-major W): same K split, N = colBase+laneM
            b.x = W[(size_t)(k + kHi)     * DD + colBase + laneM];
            b.y = W[(size_t)(k + kHi + 1) * DD + colBase + laneM];
            c = __builtin_amdgcn_wmma_f32_16x16x4_f32(
                    /*neg_a=*/false, a, /*neg_b=*/false, b,
                    /*c_mod=*/(short)0, c, /*reuse_a=*/false, /*reuse_b=*/false);
        }
        // D layout: VGPR r -> rows {r, r+8}; lanes 0-15 / 16-31 split on M+8
        #pragma unroll
        for (int r = 0; r < 8; ++r) {
            int row = rowBase + laneHi * 8 + r;
            proj[(size_t)row * DD + colBase + laneM] = c[r];
        }
    }
}

// ---------------------------------------------------------------------------
// 2) per-node attention logits el/er + global min(feat_proj) -> mu
// ---------------------------------------------------------------------------
__global__ __launch_bounds__(256) void k_node(const float* __restrict__ proj,
                                              const float* __restrict__ attn_l,
                                              const float* __restrict__ attn_r,
                                              float* __restrict__ el, float* __restrict__ er,
                                              unsigned* __restrict__ mu_enc, int N) {
    int idx = blockIdx.x * blockDim.x + threadIdx.x;   // one (node, head) each
    float lmin = INFINITY;
    if (idx < N * HH) {
        int h = idx & (HH - 1);
        const float* row = proj + (size_t)(idx >> 2) * DD + h * FF;
        const float* al = attn_l + h * FF;
        const float* ar = attn_r + h * FF;
        float sl = 0.f, sr = 0.f;
        #pragma unroll
        for (int f = 0; f < FF; ++f) {
            float v = row[f];
            sl += v * al[f];
            sr += v * ar[f];
            lmin = fminf(lmin, v);
        }
        el[idx] = sl; er[idx] = sr;
    }
    __shared__ float smin[256];
    smin[threadIdx.x] = lmin;
    __syncthreads();
    for (int off = 128; off > 0; off >>= 1) {
        if (threadIdx.x < off)
            smin[threadIdx.x] = fminf(smin[threadIdx.x], smin[threadIdx.x + off]);
        __syncthreads();
    }
    if (threadIdx.x == 0) atomicMin(mu_enc, fenc(smin[0]));
}

// ---------------------------------------------------------------------------
// 3) pre_f = pow(proj - mu + eps, p_eff), in place (once per node, not per edge)
// ---------------------------------------------------------------------------
__global__ void k_prepow(float* __restrict__ proj, const unsigned* __restrict__ mu_enc,
                         const float* __restrict__ p, const float* __restrict__ s_f, int total) {
    int i = blockIdx.x * blockDim.x + threadIdx.x;
    if (i >= total) return;
    float mu = fdec(*mu_enc);
    float pe = sigmoidf_(p[0]) + s_f[0];
    proj[i] = pow_pos(proj[i] - mu + EPSV, pe);
}

// ---------------------------------------------------------------------------
// 4) segment-max of edge logits (one thread per edge*head)
// ---------------------------------------------------------------------------
__global__ void k_edge_max(const int* __restrict__ src, const int* __restrict__ dst,
                           const float* __restrict__ el, const float* __restrict__ er,
                           unsigned* __restrict__ emax_enc, int E) {
    int idx = blockIdx.x * blockDim.x + threadIdx.x;
    if (idx >= E * HH) return;
    int e = idx >> 2, h = idx & (HH - 1);
    float v = lrelu(el[src[e] * HH + h] + er[dst[e] * HH + h]);
    atomicMax(&emax_enc[dst[e] * HH + h], fenc(v));
}

// ---------------------------------------------------------------------------
// 5) softmax denominator (one thread per edge*head)
// ---------------------------------------------------------------------------
__global__ void k_edge_denom(const int* __restrict__ src, const int* __restrict__ dst,
                             const float* __restrict__ el, const float* __restrict__ er,
                             const unsigned* __restrict__ emax_enc,
                             float* __restrict__ denom, int E) {
    int idx = blockIdx.x * blockDim.x + threadIdx.x;
    if (idx >= E * HH) return;
    int e = idx >> 2, h = idx & (HH - 1);
    int d = dst[e];
    float v = lrelu(el[src[e] * HH + h] + er[d * HH + h]);
    atomicAdd(&denom[d * HH + h], __expf(v - fdec(emax_enc[d * HH + h])));
}

// ---------------------------------------------------------------------------
// 6) weighted scatter: 32 lanes per edge, float4 per lane (L2-resident atomics)
// ---------------------------------------------------------------------------
__global__ __launch_bounds__(256) void k_edge_scatter(const int* __restrict__ src,
                                                      const int* __restrict__ dst,
                                                      const float* __restrict__ el,
                                                      const float* __restrict__ er,
                                                      const unsigned* __restrict__ emax_enc,
                                                      const float* __restrict__ denom,
                                                      const float* __restrict__ pre_f,
                                                      float* __restrict__ out, int E) {
    int idx = blockIdx.x * blockDim.x + threadIdx.x;
    int e = idx >> 5;
    if (e >= E) return;
    int lane = idx & 31;
    int h = lane >> 3;                       // 8 float4-lanes per head
    int s = src[e], d = dst[e];
    float logit = lrelu(el[s * HH + h] + er[d * HH + h]);
    float a = __expf(logit - fdec(emax_enc[d * HH + h])) / denom[d * HH + h];
    float4 v = *(const float4*)(pre_f + (size_t)s * DD + lane * 4);
    float* o = out + (size_t)d * DD + lane * 4;
    atomicAdd(o + 0, v.x * a);
    atomicAdd(o + 1, v.y * a);
    atomicAdd(o + 2, v.z * a);
    atomicAdd(o + 3, v.w * a);
}

// ---------------------------------------------------------------------------
// 7) rst = pow(pre_rst + eps, 1/p_eff) + mu + bias
// ---------------------------------------------------------------------------
__global__ void k_final(float* __restrict__ out, const float* __restrict__ bias,
                        const unsigned* __restrict__ mu_enc,
                        const float* __restrict__ p, const float* __restrict__ s_f, int total) {
    int i = blockIdx.x * blockDim.x + threadIdx.x;
    if (i >= total) return;
    float pe = sigmoidf_(p[0]) + s_f[0];
    float mu = fdec(*mu_enc);
    out[i] = pow_pos(out[i] + EPSV, 1.f / pe) + mu + bias[i & (DD - 1)];
}

// ---------------------------------------------------------------------------
extern "C" void kernel_launch(void* const* d_in, const int* in_sizes, int n_in,
                              void* d_out, int out_size, void* d_ws, size_t ws_size,
                              hipStream_t stream) {
    const float* feat   = (const float*)d_in[0];
    const int*   src    = (const int*)  d_in[1];
    const int*   dst    = (const int*)  d_in[2];
    const float* W      = (const float*)d_in[3];
    const float* attn_l = (const float*)d_in[4];
    const float* attn_r = (const float*)d_in[5];
    const float* bias   = (const float*)d_in[6];
    const float* p      = (const float*)d_in[7];
    const float* s_f    = (const float*)d_in[8];

    const int N = in_sizes[0] / DD;
    const int E = in_sizes[1];
    float* out = (float*)d_out;

    // workspace carve-out (~57.6 MB)
    char* ws = (char*)d_ws;
    float*    proj     = (float*)ws;    ws += (size_t)N * DD * sizeof(float);
    float*    el       = (float*)ws;    ws += (size_t)N * HH * sizeof(float);
    float*    er       = (float*)ws;    ws += (size_t)N * HH * sizeof(float);
    float*    denom    = (float*)ws;    ws += (size_t)N * HH * sizeof(float);
    unsigned* emax_enc = (unsigned*)ws; ws += (size_t)N * HH * sizeof(unsigned);
    unsigned* mu_enc   = (unsigned*)ws;

    const int tot = N * DD;
    k_init<<<(tot + 255) / 256, 256, 0, stream>>>(out, denom, emax_enc, mu_enc, N);
    k_gemm_wmma<<<(N + 63) / 64, 256, 0, stream>>>(feat, W, proj, N);
    k_node<<<(N * HH + 255) / 256, 256, 0, stream>>>(proj, attn_l, attn_r, el, er, mu_enc, N);
    k_prepow<<<(tot + 255) / 256, 256, 0, stream>>>(proj, mu_enc, p, s_f, tot);
    k_edge_max<<<(E * HH + 255) / 256, 256, 0, stream>>>(src, dst, el, er, emax_enc, E);
    k_edge_denom<<<(E * HH + 255) / 256, 256, 0, stream>>>(src, dst, el, er, emax_enc, denom, E);
    k_edge_scatter<<<((size_t)E * 32 + 255) / 256, 256, 0, stream>>>(src, dst, el, er, emax_enc,
                                                                     denom, proj, out, E);
    k_final<<<(tot + 255) / 256, 256, 0, stream>>>(out, bias, mu_enc, p, s_f, tot);
}